// Encoder_85830626443471
// MI455X (gfx1250) — compile-verified
//
#include <hip/hip_runtime.h>
#include <hip/hip_bf16.h>

typedef __attribute__((ext_vector_type(2))) float v2f;
typedef __attribute__((ext_vector_type(8))) float v8f;

// ---------------------------------------------------------------------------
// Constants for this problem instance
// ---------------------------------------------------------------------------
#define BB 4
#define LL 2048
#define DD 768
#define HH 12
#define II 3072
#define MM (BB * LL)          // 8192 token rows
#define QKVW (3 * DD)         // 2304

__device__ __forceinline__ v8f wmma_f32(v2f a, v2f b, v8f c) {
    return __builtin_amdgcn_wmma_f32_16x16x4_f32(false, a, false, b, (short)0, c,
                                                 false, false);
}

// CDNA5 async global->LDS copy (ASYNCcnt tracked). One instruction moves
// 16 bytes per active lane with per-lane LDS destination offsets.
__device__ __forceinline__ void async_copy_b128(const void* lds_ptr,
                                                const float* gptr) {
    // Flat shared aperture: addr[31:0] == LDS byte offset (ISA 10.2).
    unsigned lds_off = (unsigned)(size_t)lds_ptr;
    asm volatile("global_load_async_to_lds_b128 %0, %1, off"
                 :
                 : "v"(lds_off), "v"(gptr)
                 : "memory");
}

__device__ __forceinline__ void wait_async0() {
    asm volatile("s_wait_asynccnt 0x0" ::: "memory");
}

// ---------------------------------------------------------------------------
// LayerNorm: one 256-thread block per row of 768
// ---------------------------------------------------------------------------
__global__ __launch_bounds__(256) void layernorm_kernel(
    const float* __restrict__ x, const float* __restrict__ g,
    const float* __restrict__ b, float* __restrict__ y) {
    __shared__ float red[256];
    const int row = blockIdx.x;
    const int tid = threadIdx.x;
    const float* xr = x + (size_t)row * DD;
    float vals[3];
    float s = 0.f, s2 = 0.f;
#pragma unroll
    for (int i = 0; i < 3; ++i) {
        float v = xr[tid + i * 256];
        vals[i] = v;
        s += v;
        s2 += v * v;
    }
    red[tid] = s;
    __syncthreads();
    for (int st = 128; st > 0; st >>= 1) {
        if (tid < st) red[tid] += red[tid + st];
        __syncthreads();
    }
    const float mean = red[0] * (1.0f / DD);
    __syncthreads();
    red[tid] = s2;
    __syncthreads();
    for (int st = 128; st > 0; st >>= 1) {
        if (tid < st) red[tid] += red[tid + st];
        __syncthreads();
    }
    const float var = red[0] * (1.0f / DD) - mean * mean;
    const float rstd = rsqrtf(var + 1e-5f);
    float* yr = y + (size_t)row * DD;
#pragma unroll
    for (int i = 0; i < 3; ++i) {
        const int c = tid + i * 256;
        yr[c] = (vals[i] - mean) * rstd * g[c] + b[c];
    }
}

// ---------------------------------------------------------------------------
// WMMA GEMM:  C[M x N] = A[M x K] @ W[K x N] + bias  (+ epilogue)
//   EPI = 0 : none;  EPI = 1 : += res;  EPI = 2 : C = res + gelu_exact(val)
// Block tile 128(M) x 128(N), BK = 16; 8 waves (4x2), wave tile 32x64.
// Double-buffered LDS filled by async global->LDS DMA.
// ---------------------------------------------------------------------------
template <int EPI>
__global__ __launch_bounds__(256) void gemm_kernel(
    const float* __restrict__ A, const float* __restrict__ W,
    const float* __restrict__ bias, const float* __restrict__ res,
    float* __restrict__ C, int M, int N, int K) {
    __shared__ float As[2][128][20];   // stride 20: 16B-aligned, conflict-free
    __shared__ float Bs[2][16][136];   // stride 136: halves hit disjoint banks

    const int tid  = threadIdx.x;
    const int lane = tid & 31;
    const int wave = tid >> 5;
    const int wm   = wave >> 1;  // 0..3 (M)
    const int wn   = wave & 1;   // 0..1 (N)
    const int bm   = blockIdx.y * 128;
    const int bn   = blockIdx.x * 128;
    const int lm   = lane & 15;
    const int kh   = lane >> 4;  // K-pair half select

    // Per-thread async segment coordinates (2 x b128 for A, 2 x b128 for B)
    const int ar0 = tid >> 2;             // A rows 0..63 / 64..127
    const int ac4 = (tid & 3) * 4;        // A col group
    const int bk0 = tid >> 5;             // B k-rows 0..7 / 8..15
    const int bc4 = (tid & 31) * 4;       // B col group

    v8f acc[2][4] = {};

    auto issue_tile = [&](int buf, int kb) {
        async_copy_b128(&As[buf][ar0][ac4],
                        A + (size_t)(bm + ar0) * K + kb + ac4);
        async_copy_b128(&As[buf][ar0 + 64][ac4],
                        A + (size_t)(bm + ar0 + 64) * K + kb + ac4);
        async_copy_b128(&Bs[buf][bk0][bc4],
                        W + (size_t)(kb + bk0) * N + bn + bc4);
        async_copy_b128(&Bs[buf][bk0 + 8][bc4],
                        W + (size_t)(kb + bk0 + 8) * N + bn + bc4);
    };

    int cur = 0;
    issue_tile(0, 0);

    for (int kb = 0; kb < K; kb += 16) {
        wait_async0();        // my DMA into As/Bs[cur] landed in LDS
        __syncthreads();      // everyone's DMA landed; prev compute finished
        const int nxt = cur ^ 1;
        if (kb + 16 < K) issue_tile(nxt, kb + 16);  // prefetch next chunk
        // ---- compute: 4 k-steps x 2x4 subtiles = 32 WMMA ----
        const float(*Ac)[20]  = As[cur];
        const float(*Bc)[136] = Bs[cur];
#pragma unroll
        for (int ks = 0; ks < 16; ks += 4) {
            v2f af[2], bf[4];
#pragma unroll
            for (int i = 0; i < 2; ++i)
                af[i] = *(const v2f*)&Ac[wm * 32 + i * 16 + lm][ks + 2 * kh];
#pragma unroll
            for (int j = 0; j < 4; ++j) {
                const int col = wn * 64 + j * 16 + lm;
                v2f t;
                t.x = Bc[ks + 2 * kh][col];
                t.y = Bc[ks + 2 * kh + 1][col];
                bf[j] = t;
            }
#pragma unroll
            for (int i = 0; i < 2; ++i)
#pragma unroll
                for (int j = 0; j < 4; ++j)
                    acc[i][j] = wmma_f32(af[i], bf[j], acc[i][j]);
        }
        cur = nxt;
    }

    // ---- epilogue ----
#pragma unroll
    for (int i = 0; i < 2; ++i) {
#pragma unroll
        for (int j = 0; j < 4; ++j) {
            const int col = bn + wn * 64 + j * 16 + lm;
            const float bv = bias[col];
#pragma unroll
            for (int r = 0; r < 8; ++r) {
                const int row = bm + wm * 32 + i * 16 + kh * 8 + r;
                const size_t idx = (size_t)row * N + col;
                float v = acc[i][j][r] + bv;
                if (EPI == 1) {
                    v += res[idx];
                } else if (EPI == 2) {
                    const float ge =
                        0.5f * v * (1.0f + erff(v * 0.70710678118654752f));
                    v = res[idx] + ge;
                }
                C[idx] = v;
            }
        }
    }
}

// ---------------------------------------------------------------------------
// Flash attention: grid (L/64, B*H), 128 threads = 4 waves, each wave owns a
// 16-query-row tile of one head. qkv rows are 2304 wide: head h occupies
// [h*192, h*192+192) as q|k|v of 64 each. Output written as (b,l, h*64+d).
// ---------------------------------------------------------------------------
__global__ __launch_bounds__(128) void attention_kernel(
    const float* __restrict__ qkv, const unsigned char* __restrict__ mask,
    float* __restrict__ out) {
    const int bh   = blockIdx.y;
    const int b    = bh / HH;
    const int h    = bh % HH;
    const int wave = threadIdx.x >> 5;
    const int lane = threadIdx.x & 31;
    const int lm   = lane & 15;
    const int kh   = lane >> 4;
    const int i0   = (blockIdx.x * 4 + wave) * 16;  // query row base

    __shared__ float SbufA[4][16][18];
    __shared__ float AlphaA[4][16];
    __shared__ float LrowA[4][16];
    float(*Sbuf)[18] = SbufA[wave];
    float* Alpha     = AlphaA[wave];
    float* Lrow      = LrowA[wave];

    // preload Q fragments for all 16 K-steps (head dim 64)
    const size_t rowQ = (size_t)(b * LL + i0 + lm) * QKVW + h * 192;
    v2f qf[16];
#pragma unroll
    for (int ks = 0; ks < 16; ++ks)
        qf[ks] = *(const v2f*)(qkv + rowQ + ks * 4 + 2 * kh);

    v8f o[4] = {};
    float m_i = -__builtin_inff();  // owned by lanes 0..15 (row = lane)
    float l_i = 0.f;
    const float scale = 0.125f;  // 1/sqrt(64)
    const unsigned char* mrow = mask + (size_t)b * LL;

    for (int j0 = 0; j0 < LL; j0 += 16) {
        // ---- S = Q @ K^T (16x16), chained over K=64 ----
        v8f s = {};
        const size_t rowK = (size_t)(b * LL + j0 + lm) * QKVW + h * 192 + 64;
#pragma unroll
        for (int ks = 0; ks < 16; ++ks) {
            const v2f kf = *(const v2f*)(qkv + rowK + ks * 4 + 2 * kh);
            s = wmma_f32(qf[ks], kf, s);
        }
        // ---- mask + scale, spill to LDS in (row, col) form ----
        const bool mv = mrow[j0 + lm] != 0;
#pragma unroll
        for (int r = 0; r < 8; ++r)
            Sbuf[kh * 8 + r][lm] = mv ? s[r] * scale : -__builtin_inff();
        __syncthreads();
        // ---- online softmax update (lanes 0..15, one row each) ----
        if (lane < 16) {
            float mx = m_i;
#pragma unroll
            for (int j = 0; j < 16; ++j) mx = fmaxf(mx, Sbuf[lane][j]);
            float alpha;
            float sum = 0.f;
            if (mx == -__builtin_inff()) {
                alpha = 1.f;
#pragma unroll
                for (int j = 0; j < 16; ++j) Sbuf[lane][j] = 0.f;
            } else {
                alpha = __expf(m_i - mx);
#pragma unroll
                for (int j = 0; j < 16; ++j) {
                    const float p = __expf(Sbuf[lane][j] - mx);
                    Sbuf[lane][j] = p;
                    sum += p;
                }
            }
            l_i = l_i * alpha + sum;
            m_i = mx;
            Alpha[lane] = alpha;
        }
        __syncthreads();
        // ---- rescale running O by alpha(row) ----
#pragma unroll
        for (int t = 0; t < 4; ++t)
#pragma unroll
            for (int r = 0; r < 8; ++r) o[t][r] *= Alpha[kh * 8 + r];
        // ---- O += P @ V  (K=16 over this key block, N=64 in 4 tiles) ----
#pragma unroll
        for (int ks = 0; ks < 4; ++ks) {
            const v2f pf = *(const v2f*)&Sbuf[lm][ks * 4 + 2 * kh];
            const size_t rowV0 =
                (size_t)(b * LL + j0 + ks * 4 + 2 * kh) * QKVW + h * 192 + 128;
            const size_t rowV1 = rowV0 + QKVW;
#pragma unroll
            for (int t = 0; t < 4; ++t) {
                v2f vf;
                vf.x = qkv[rowV0 + t * 16 + lm];
                vf.y = qkv[rowV1 + t * 16 + lm];
                o[t] = wmma_f32(pf, vf, o[t]);
            }
        }
        __syncthreads();  // Sbuf/Alpha reused next iteration
    }
    // ---- finalize: divide by row sums, write (b, l, h*64 + d) ----
    if (lane < 16) Lrow[lane] = l_i;
    __syncthreads();
#pragma unroll
    for (int t = 0; t < 4; ++t) {
#pragma unroll
        for (int r = 0; r < 8; ++r) {
            const int row = kh * 8 + r;
            const float val = o[t][r] / Lrow[row];
            out[(size_t)(b * LL + i0 + row) * DD + h * 64 + t * 16 + lm] = val;
        }
    }
}

// ---------------------------------------------------------------------------
// Launch: full encoder block pipeline
// ---------------------------------------------------------------------------
extern "C" void kernel_launch(void* const* d_in, const int* in_sizes, int n_in,
                              void* d_out, int out_size, void* d_ws,
                              size_t ws_size, hipStream_t stream) {
    const float* x            = (const float*)d_in[0];
    const unsigned char* am   = (const unsigned char*)d_in[1];
    const float* ln1_g        = (const float*)d_in[2];
    const float* ln1_b        = (const float*)d_in[3];
    const float* Wqkv         = (const float*)d_in[4];
    const float* bqkv         = (const float*)d_in[5];
    const float* Wo           = (const float*)d_in[6];
    const float* bo           = (const float*)d_in[7];
    const float* ln2_g        = (const float*)d_in[8];
    const float* ln2_b        = (const float*)d_in[9];
    const float* W1           = (const float*)d_in[10];
    const float* b1           = (const float*)d_in[11];
    const float* W2           = (const float*)d_in[12];
    const float* b2           = (const float*)d_in[13];
    float* out                = (float*)d_out;

    float* ws   = (float*)d_ws;
    float* xn   = ws;                          // 8192*768   (reused for ln2 out)
    float* qkv  = xn + (size_t)MM * DD;        // 8192*2304
    float* attn = qkv + (size_t)MM * QKVW;     // 8192*768
    float* res1 = attn + (size_t)MM * DD;      // 8192*768
    float* ffn1 = res1 + (size_t)MM * DD;      // 8192*3072

    // 1) LN1
    layernorm_kernel<<<MM, 256, 0, stream>>>(x, ln1_g, ln1_b, xn);
    // 2) qkv = xn @ Wqkv + bqkv
    gemm_kernel<0><<<dim3(QKVW / 128, MM / 128), 256, 0, stream>>>(
        xn, Wqkv, bqkv, nullptr, qkv, MM, QKVW, DD);
    // 3) attention
    attention_kernel<<<dim3(LL / 64, BB * HH), 128, 0, stream>>>(qkv, am, attn);
    // 4) res1 = attn @ Wo + bo + x
    gemm_kernel<1><<<dim3(DD / 128, MM / 128), 256, 0, stream>>>(
        attn, Wo, bo, x, res1, MM, DD, DD);
    // 5) LN2 -> xn (reuse)
    layernorm_kernel<<<MM, 256, 0, stream>>>(res1, ln2_g, ln2_b, xn);
    // 6) ffn1 = xn @ W1 + b1
    gemm_kernel<0><<<dim3(II / 128, MM / 128), 256, 0, stream>>>(
        xn, W1, b1, nullptr, ffn1, MM, II, DD);
    // 7) out = res1 + gelu(ffn1 @ W2 + b2)
    gemm_kernel<2><<<dim3(DD / 128, MM / 128), 256, 0, stream>>>(
        ffn1, W2, b2, res1, out, MM, DD, II);
}